// BatchWhiten_54099408061166
// MI455X (gfx1250) — compile-verified
//
#include <hip/hip_runtime.h>
#include <math.h>
#include <stdint.h>

typedef float v2f __attribute__((ext_vector_type(2)));
typedef float v8f __attribute__((ext_vector_type(8)));

#define MOMENTUM 0.1f
#define NS_ITERS 18

__device__ __forceinline__ v8f wmma_f32(v2f a, v2f b, v8f c) {
  // D = A(16x4) * B(4x16) + C, fp32
  return __builtin_amdgcn_wmma_f32_16x16x4_f32(false, a, false, b, (short)0, c,
                                               false, false);
}

// ---------------- kernel 0: zero the covariance accumulator ----------------
__global__ void bw_zero(float* __restrict__ cov) {
  cov[blockIdx.x * 256 + threadIdx.x] = 0.0f;
}

// ---------------- kernel 1: cov += x^T x (partial, atomic) ----------------
// Double-buffered chunks staged with CDNA5 async memory->LDS copies.
// LDS holds a chunk transposed: lds[f * K1_S + s]; K1_S even keeps the
// ds_load_b64 fragment reads 8B-aligned; K1_S=98 spreads banks.
#define K1_R 96
#define K1_S 98
#define K1_ELEMS_PER_THREAD ((K1_R * 64) / 256) /* 24 async b32 per lane */

// One wave32 instruction per call: per-lane global src -> per-lane LDS dest.
// Flat pointers into LDS carry the LDS byte offset in their low 32 bits
__device__ __forceinline__ void async_g2l_b32(float* lds_dst,
                                              const float* gsrc) {
  unsigned lds_off = (unsigned)(uintptr_t)lds_dst;
  asm volatile("global_load_async_to_lds_b32 %0, %1, off"
               :
               : "v"(lds_off), "v"(gsrc)
               : "memory");
}

__device__ __forceinline__ void issue_chunk_async(float* buf,
                                                  const float* __restrict__ x,
                                                  long c, int tid) {
  for (int i = 0; i < K1_ELEMS_PER_THREAD; ++i) {
    int e = tid + 256 * i;
    int s = e >> 6, f = e & 63;
    async_g2l_b32(buf + f * K1_S + s, x + (c * K1_R + s) * 64 + f);
  }
}

__global__ void bw_covar(const float* __restrict__ x, float* __restrict__ cov,
                         long N) {
  __shared__ float lds[2][64 * K1_S];
  const int tid = threadIdx.x;
  const int lane = tid & 31;
  const int w = tid >> 5;          // wave 0..7, owns output tiles 2w, 2w+1
  const int fm = w >> 1;           // tile row (feature block of A)
  const int fn0 = (2 * w) & 3;     // tile cols (feature blocks of B)
  const int fn1 = fn0 + 1;
  const int m = lane & 15;
  const int kk = (lane >> 4) << 1; // k sub-offset per half-wave

  v8f acc0 = {0, 0, 0, 0, 0, 0, 0, 0};
  v8f acc1 = {0, 0, 0, 0, 0, 0, 0, 0};

  const long nchunks = (N + K1_R - 1) / K1_R;
  const long cstride = gridDim.x;
  int par = 0;

  const long c0 = blockIdx.x;
  if (c0 < nchunks && (c0 + 1) * K1_R <= N)
    issue_chunk_async(&lds[0][0], x, c0, tid); // prefetch first chunk

  for (long c = c0; c < nchunks; c += cstride) {
    const long nxt = c + cstride;
    const bool cur_full = ((c + 1) * K1_R <= N);
    const bool nxt_full = (nxt < nchunks) && ((nxt + 1) * K1_R <= N);

    if (nxt_full) issue_chunk_async(&lds[par ^ 1][0], x, nxt, tid);

    if (cur_full) {
      // async loads complete in order: waiting down to the next chunk's
      // in-flight count guarantees the current chunk has landed in LDS.
      if (nxt_full)
        asm volatile("s_wait_asynccnt 0x18" ::: "memory"); // <= 24
      else
        asm volatile("s_wait_asynccnt 0x0" ::: "memory");
    } else {
      // ragged tail chunk (at most one): synchronous zero-padded fill
      for (int e = tid; e < K1_R * 64; e += 256) {
        int s = e >> 6, f = e & 63;
        long row = c * (long)K1_R + s;
        lds[par][f * K1_S + s] = (row < N) ? x[row * 64 + f] : 0.0f;
      }
    }
    __syncthreads();

    const float* buf = &lds[par][0];
#pragma unroll 4
    for (int ks = 0; ks < K1_R / 4; ++ks) {
      int kb = ks * 4 + kk;
      // identical lane layout serves as both A (16x4) and B (4x16) fragment
      v2f a  = *(const v2f*)(buf + (16 * fm  + m) * K1_S + kb);
      v2f b0 = *(const v2f*)(buf + (16 * fn0 + m) * K1_S + kb);
      v2f b1 = *(const v2f*)(buf + (16 * fn1 + m) * K1_S + kb);
      acc0 = wmma_f32(a, b0, acc0);
      acc1 = wmma_f32(a, b1, acc1);
    }
    __syncthreads();
    par ^= 1;
  }

  const int nh = lane >> 4;
#pragma unroll
  for (int r = 0; r < 8; ++r) {
    int row = 16 * fm + r + 8 * nh; // C/D layout: M = r + 8*(lane>=16)
    atomicAdd(&cov[row * 64 + 16 * fn0 + m], acc0[r]);
    atomicAdd(&cov[row * 64 + 16 * fn1 + m], acc1[r]);
  }
}

// ------- kernel 2: EMA + Newton-Schulz inverse matrix square root ---------
// Single workgroup, 256 threads, WMMA for all 64x64 matmuls.
#define LSTR 66
__global__ void bw_invsqrt(const float* __restrict__ cov,
                           const float* __restrict__ running,
                           float* __restrict__ Bout, long N) {
  __shared__ float Y[64 * LSTR];
  __shared__ float Z[64 * LSTR];
  __shared__ float T[64 * LSTR];
  __shared__ float s_c;
  const int tid = threadIdx.x;
  const int lane = tid & 31;
  const int w = tid >> 5;
  const int fm = w >> 1;
  const int fn0 = (2 * w) & 3;
  const int fn1 = fn0 + 1;
  const int m = lane & 15;
  const int kk = (lane >> 4) << 1;
  const int nh = lane >> 4;

  // C = mom*running + (1-mom)*covSum/N ; Frobenius norm^2 reduction
  const float invN = 1.0f / (float)N;
  float ss = 0.0f;
#pragma unroll
  for (int j = 0; j < 16; ++j) {
    int idx = tid * 16 + j;
    int row = idx >> 6, col = idx & 63;
    float cv = MOMENTUM * running[idx] + (1.0f - MOMENTUM) * cov[idx] * invN;
    Y[row * LSTR + col] = cv;
    ss += cv * cv;
  }
  T[tid] = ss; // reuse T as reduction scratch
  __syncthreads();
  for (int s = 128; s > 0; s >>= 1) {
    if (tid < s) T[tid] += T[tid + s];
    __syncthreads();
  }
  if (tid == 0) s_c = sqrtf(T[0]); // c = ||C||_F  (>= spectral norm)
  __syncthreads();
  const float invc = 1.0f / s_c;
#pragma unroll
  for (int j = 0; j < 16; ++j) {
    int idx = tid * 16 + j;
    int row = idx >> 6, col = idx & 63;
    Y[row * LSTR + col] *= invc;                      // Y0 = C/c
    Z[row * LSTR + col] = (row == col) ? 1.0f : 0.0f; // Z0 = I
  }
  __syncthreads();

  for (int it = 0; it < NS_ITERS; ++it) {
    // T = 1.5*I - 0.5*(Z @ Y)
    v8f t0 = {0, 0, 0, 0, 0, 0, 0, 0};
    v8f t1 = {0, 0, 0, 0, 0, 0, 0, 0};
#pragma unroll
    for (int ks = 0; ks < 16; ++ks) {
      int kb = ks * 4 + kk;
      v2f a = *(const v2f*)(Z + (16 * fm + m) * LSTR + kb);
      v2f b0, b1;
      b0.x = Y[(kb + 0) * LSTR + 16 * fn0 + m];
      b0.y = Y[(kb + 1) * LSTR + 16 * fn0 + m];
      b1.x = Y[(kb + 0) * LSTR + 16 * fn1 + m];
      b1.y = Y[(kb + 1) * LSTR + 16 * fn1 + m];
      t0 = wmma_f32(a, b0, t0);
      t1 = wmma_f32(a, b1, t1);
    }
#pragma unroll
    for (int r = 0; r < 8; ++r) {
      int row = 16 * fm + r + 8 * nh;
      int c0 = 16 * fn0 + m, c1 = 16 * fn1 + m;
      T[row * LSTR + c0] = ((row == c0) ? 1.5f : 0.0f) - 0.5f * t0[r];
      T[row * LSTR + c1] = ((row == c1) ? 1.5f : 0.0f) - 0.5f * t1[r];
    }
    __syncthreads();
    // Ynew = Y @ T, Znew = T @ Z  (accumulate in registers, then commit)
    v8f y0 = {0, 0, 0, 0, 0, 0, 0, 0};
    v8f y1 = {0, 0, 0, 0, 0, 0, 0, 0};
    v8f z0 = {0, 0, 0, 0, 0, 0, 0, 0};
    v8f z1 = {0, 0, 0, 0, 0, 0, 0, 0};
#pragma unroll
    for (int ks = 0; ks < 16; ++ks) {
      int kb = ks * 4 + kk;
      v2f aY = *(const v2f*)(Y + (16 * fm + m) * LSTR + kb);
      v2f aT = *(const v2f*)(T + (16 * fm + m) * LSTR + kb);
      v2f bT0, bT1, bZ0, bZ1;
      bT0.x = T[(kb + 0) * LSTR + 16 * fn0 + m];
      bT0.y = T[(kb + 1) * LSTR + 16 * fn0 + m];
      bT1.x = T[(kb + 0) * LSTR + 16 * fn1 + m];
      bT1.y = T[(kb + 1) * LSTR + 16 * fn1 + m];
      bZ0.x = Z[(kb + 0) * LSTR + 16 * fn0 + m];
      bZ0.y = Z[(kb + 1) * LSTR + 16 * fn0 + m];
      bZ1.x = Z[(kb + 0) * LSTR + 16 * fn1 + m];
      bZ1.y = Z[(kb + 1) * LSTR + 16 * fn1 + m];
      y0 = wmma_f32(aY, bT0, y0);
      y1 = wmma_f32(aY, bT1, y1);
      z0 = wmma_f32(aT, bZ0, z0);
      z1 = wmma_f32(aT, bZ1, z1);
    }
    __syncthreads(); // everyone done reading Y/Z/T
#pragma unroll
    for (int r = 0; r < 8; ++r) {
      int row = 16 * fm + r + 8 * nh;
      Y[row * LSTR + 16 * fn0 + m] = y0[r];
      Y[row * LSTR + 16 * fn1 + m] = y1[r];
      Z[row * LSTR + 16 * fn0 + m] = z0[r];
      Z[row * LSTR + 16 * fn1 + m] = z1[r];
    }
    __syncthreads();
  }

  // B = Z / sqrt(c)
  const float sc = rsqrtf(s_c);
#pragma unroll
  for (int j = 0; j < 16; ++j) {
    int idx = tid * 16 + j;
    int row = idx >> 6, col = idx & 63;
    Bout[idx] = Z[row * LSTR + col] * sc;
  }
}

// ---------------- kernel 3: out = x @ B ----------------
// B fragments (64 of them, 128 floats) live in registers for the whole kernel.
// x read (last use) and out write are nontemporal: pure streaming, keep L2 free.
__global__ __launch_bounds__(256) void bw_apply(const float* __restrict__ x,
                                                const float* __restrict__ Bm,
                                                float* __restrict__ out,
                                                long N) {
  const int tid = threadIdx.x;
  const int lane = tid & 31;
  const int m = lane & 15;
  const int kk = (lane >> 4) << 1;
  const int nh = lane >> 4;

  v2f bf[16][4];
#pragma unroll
  for (int ks = 0; ks < 16; ++ks) {
    int k0 = ks * 4 + kk;
#pragma unroll
    for (int fn = 0; fn < 4; ++fn) {
      v2f b;
      b.x = Bm[(k0 + 0) * 64 + 16 * fn + m];
      b.y = Bm[(k0 + 1) * 64 + 16 * fn + m];
      bf[ks][fn] = b;
    }
  }

  const long ntiles = (N + 15) / 16;
  const long gw = (long)blockIdx.x * 8 + (tid >> 5);
  const long stride = (long)gridDim.x * 8;
  const v8f zero8 = {0, 0, 0, 0, 0, 0, 0, 0};

  for (long t = gw; t < ntiles; t += stride) {
    const long n0 = t * 16;
    v8f acc0 = zero8, acc1 = zero8, acc2 = zero8, acc3 = zero8;
    long rowA = n0 + m;
    if (rowA > N - 1) rowA = N - 1; // clamp loads on ragged tail
    const float* xrow = x + rowA * 64 + kk;
#pragma unroll
    for (int ks = 0; ks < 16; ++ks) {
      v2f a = __builtin_nontemporal_load((const v2f*)(xrow + ks * 4));
      acc0 = wmma_f32(a, bf[ks][0], acc0);
      acc1 = wmma_f32(a, bf[ks][1], acc1);
      acc2 = wmma_f32(a, bf[ks][2], acc2);
      acc3 = wmma_f32(a, bf[ks][3], acc3);
    }
#pragma unroll
    for (int r = 0; r < 8; ++r) {
      long row = n0 + r + 8 * nh;
      if (row < N) {
        float* orow = out + row * 64 + m;
        __builtin_nontemporal_store(acc0[r], orow + 0);
        __builtin_nontemporal_store(acc1[r], orow + 16);
        __builtin_nontemporal_store(acc2[r], orow + 32);
        __builtin_nontemporal_store(acc3[r], orow + 48);
      }
    }
  }
}

extern "C" void kernel_launch(void* const* d_in, const int* in_sizes, int n_in,
                              void* d_out, int out_size, void* d_ws,
                              size_t ws_size, hipStream_t stream) {
  const float* x = (const float*)d_in[0];        // [N,64] fp32
  const float* running = (const float*)d_in[1];  // [64,64] fp32
  float* out = (float*)d_out;                    // [N,64] fp32
  float* cov = (float*)d_ws;                     // 4096 floats: sum x^T x
  float* Bm = cov + 4096;                        // 4096 floats: whitening B
  const long N = (long)in_sizes[0] / 64;

  bw_zero<<<16, 256, 0, stream>>>(cov);
  bw_covar<<<512, 256, 0, stream>>>(x, cov, N);
  bw_invsqrt<<<1, 256, 0, stream>>>(cov, running, Bm, N);
  bw_apply<<<1024, 256, 0, stream>>>(x, Bm, out, N);
}